// simclr_78563541778990
// MI455X (gfx1250) — compile-verified
//
#include <hip/hip_runtime.h>
#include <hip/hip_bf16.h>

#define N_TOT 8192
#define DIM   512
#define INV_T 5.0f            // 1 / 0.2
#define LDA   514             // padded LDS stride (floats) to dodge bank conflicts
#define ROWS  64              // rows per workgroup (A block = 64x512 f32 = 128KB LDS)

typedef float v2f __attribute__((ext_vector_type(2)));
typedef float v8f __attribute__((ext_vector_type(8)));

// ---------------------------------------------------------------- normalize
__global__ void k_normalize(const float* __restrict__ z, float* __restrict__ zn) {
    int gw   = (blockIdx.x * blockDim.x + threadIdx.x) >> 5;   // one wave per row
    int lane = threadIdx.x & 31;
    if (gw >= N_TOT) return;
    const float* row = z + (size_t)gw * DIM;
    float ss = 0.0f;
    for (int c = lane; c < DIM; c += 32) { float v = row[c]; ss += v * v; }
    #pragma unroll
    for (int m = 16; m >= 1; m >>= 1) ss += __shfl_xor(ss, m, 32);
    float sc = 1.0f / fmaxf(sqrtf(ss), 1e-12f);
    float* orow = zn + (size_t)gw * DIM;
    for (int c = lane; c < DIM; c += 32) orow[c] = row[c] * sc;
}

// ---------------------------------------------------------------- init accumulators
__global__ void k_init(float* __restrict__ accum) {
    if (threadIdx.x < 2) accum[threadIdx.x] = 0.0f;
}

// ---------------------------------------------------------------- GEMM pass 1: row stats
__global__ void __launch_bounds__(256)
k_stats(const float* __restrict__ zn, float* __restrict__ S1, float* __restrict__ S2) {
    __shared__ float ldsA[ROWS * LDA];
    __shared__ float s1s[ROWS], s2s[ROWS];

    const int tid     = threadIdx.x;
    const int rowbase = blockIdx.x * ROWS;

    // stage ROWS x 512 A block into LDS as float2
    for (int idx = tid; idx < ROWS * (DIM / 2); idx += 256) {
        int r  = idx >> 8;            // DIM/2 == 256 float2 per row
        int c2 = idx & 255;
        v2f v = *(const v2f*)(zn + (size_t)(rowbase + r) * DIM + c2 * 2);
        *(v2f*)(&ldsA[r * LDA + c2 * 2]) = v;
    }
    for (int i = tid; i < ROWS; i += 256) { s1s[i] = 0.0f; s2s[i] = 0.0f; }
    __syncthreads();

    const int wave = tid >> 5, lane = tid & 31;
    const int lo = lane & 15, hi = lane >> 4, koff = hi * 2;
    const int rt = wave & 3;          // row tile within block  (4 tiles of 16 rows)
    const int ch = wave >> 2;         // column half (0/1), 256 column tiles each
    const float* Ap = &ldsA[(rt * 16 + lo) * LDA + koff];

    float p1[8], p2[8];
    #pragma unroll
    for (int v = 0; v < 8; ++v) { p1[v] = 0.0f; p2[v] = 0.0f; }

    const v8f zacc = {0.f,0.f,0.f,0.f,0.f,0.f,0.f,0.f};

    for (int it = 0; it < 64; ++it) {
        const int cb = (ch * 256 + it * 4) * 16;      // first of 4 16-wide column tiles
        const float* Bp[4];
        v8f acc[4];
        #pragma unroll
        for (int j = 0; j < 4; ++j) {
            acc[j] = zacc;
            Bp[j]  = zn + (size_t)(cb + 16 * j + lo) * DIM + koff;
        }
        #pragma unroll 4
        for (int k0 = 0; k0 < DIM; k0 += 4) {
            v2f a = *(const v2f*)(Ap + k0);
            #pragma unroll
            for (int j = 0; j < 4; ++j) {
                v2f b = *(const v2f*)(Bp[j] + k0);
                acc[j] = __builtin_amdgcn_wmma_f32_16x16x4_f32(
                    false, a, false, b, (short)0, acc[j], false, false);
            }
        }
        #pragma unroll
        for (int v = 0; v < 8; ++v) {
            #pragma unroll
            for (int j = 0; j < 4; ++j) {
                float s = acc[j][v];
                p1[v] += s;
                p2[v] += s * s;
            }
        }
    }
    #pragma unroll
    for (int v = 0; v < 8; ++v) {
        atomicAdd(&s1s[rt * 16 + hi * 8 + v], p1[v]);
        atomicAdd(&s2s[rt * 16 + hi * 8 + v], p2[v]);
    }
    __syncthreads();
    for (int i = tid; i < ROWS; i += 256) {
        S1[rowbase + i] = s1s[i];
        S2[rowbase + i] = s2s[i];
    }
}

// ---------------------------------------------------------------- finalize stats
__global__ void k_stat_fin(const float* __restrict__ S1, const float* __restrict__ S2,
                           float* __restrict__ mu, float* __restrict__ i2v) {
    int i = blockIdx.x * blockDim.x + threadIdx.x;
    if (i >= N_TOT) return;
    float s1 = S1[i], s2 = S2[i];
    const float n = (float)N_TOT;
    float m     = 1.0f - s1 / n;                       // mean of dist row i
    float sumsq = n - 2.0f * s1 + s2;                  // sum dist^2 over row i
    float var   = (sumsq - n * m * m) / (n - 1.0f);    // unbiased (ddof=1)
    mu[i]  = m;
    i2v[i] = 1.0f / (2.0f * var);
}

// ---------------------------------------------------------------- GEMM pass 2: neg sum
__global__ void __launch_bounds__(256)
k_neg(const float* __restrict__ zn, const int* __restrict__ labels,
      const float* __restrict__ mu, const float* __restrict__ i2v,
      float* __restrict__ negAcc) {
    __shared__ float ldsA[ROWS * LDA];
    __shared__ int   labR[ROWS];

    const int tid     = threadIdx.x;
    const int rowbase = blockIdx.x * ROWS;

    for (int idx = tid; idx < ROWS * (DIM / 2); idx += 256) {
        int r  = idx >> 8;
        int c2 = idx & 255;
        v2f v = *(const v2f*)(zn + (size_t)(rowbase + r) * DIM + c2 * 2);
        *(v2f*)(&ldsA[r * LDA + c2 * 2]) = v;
    }
    for (int i = tid; i < ROWS; i += 256) labR[i] = labels[rowbase + i];
    __syncthreads();

    const int wave = tid >> 5, lane = tid & 31;
    const int lo = lane & 15, hi = lane >> 4, koff = hi * 2;
    const int rt = wave & 3;
    const int ch = wave >> 2;
    const float* Ap = &ldsA[(rt * 16 + lo) * LDA + koff];

    const v8f zacc = {0.f,0.f,0.f,0.f,0.f,0.f,0.f,0.f};
    float local = 0.0f;

    for (int it = 0; it < 64; ++it) {
        const int cb = (ch * 256 + it * 4) * 16;
        const float* Bp[4];
        v8f acc[4];
        #pragma unroll
        for (int j = 0; j < 4; ++j) {
            acc[j] = zacc;
            Bp[j]  = zn + (size_t)(cb + 16 * j + lo) * DIM + koff;
        }
        #pragma unroll 4
        for (int k0 = 0; k0 < DIM; k0 += 4) {
            v2f a = *(const v2f*)(Ap + k0);
            #pragma unroll
            for (int j = 0; j < 4; ++j) {
                v2f b = *(const v2f*)(Bp[j] + k0);
                acc[j] = __builtin_amdgcn_wmma_f32_16x16x4_f32(
                    false, a, false, b, (short)0, acc[j], false, false);
            }
        }
        // epilogue: exp(s/T) * exp((d - mu_col)^2 / (2 var_col)), masked
        #pragma unroll
        for (int j = 0; j < 4; ++j) {
            const int col = cb + 16 * j + lo;        // column index for this lane
            const float muc = mu[col];
            const float ivc = i2v[col];
            const int   lc  = labels[col];
            #pragma unroll
            for (int v = 0; v < 8; ++v) {
                const int row = rowbase + rt * 16 + hi * 8 + v;
                if (row != col && labR[rt * 16 + hi * 8 + v] != lc) {
                    float s = acc[j][v];
                    float d = 1.0f - s;
                    local += __expf(s * INV_T + (d - muc) * (d - muc) * ivc);
                }
            }
        }
    }
    #pragma unroll
    for (int m = 16; m >= 1; m >>= 1) local += __shfl_xor(local, m, 32);
    if (lane == 0) atomicAdd(negAcc, local);
}

// ---------------------------------------------------------------- positive pairs
__global__ void k_pos(const float* __restrict__ zn, const int* __restrict__ labels,
                      float* __restrict__ posAcc) {
    const int B = N_TOT / 2;
    int gw   = (blockIdx.x * blockDim.x + threadIdx.x) >> 5;  // one wave per pair
    int lane = threadIdx.x & 31;
    if (gw >= B) return;
    if (labels[gw] != labels[gw + B]) return;                 // wave-uniform
    const float* r0 = zn + (size_t)gw * DIM;
    const float* r1 = zn + (size_t)(gw + B) * DIM;
    float d = 0.0f;
    for (int c = lane; c < DIM; c += 32) d += r0[c] * r1[c];
    #pragma unroll
    for (int m = 16; m >= 1; m >>= 1) d += __shfl_xor(d, m, 32);
    if (lane == 0) atomicAdd(posAcc, __expf(d * INV_T));
}

// ---------------------------------------------------------------- final scalar
__global__ void k_final(const float* __restrict__ accum, float* __restrict__ out) {
    float neg = accum[0], pos = accum[1];
    out[0] = -logf(pos / (pos + neg));
}

// ---------------------------------------------------------------- launcher
extern "C" void kernel_launch(void* const* d_in, const int* in_sizes, int n_in,
                              void* d_out, int out_size, void* d_ws, size_t ws_size,
                              hipStream_t stream) {
    const float* z      = (const float*)d_in[0];
    const int*   labels = (const int*)d_in[1];
    float*       out    = (float*)d_out;

    float* ws    = (float*)d_ws;
    float* zn    = ws;                                 // N*D floats (16 MB)
    float* S1    = ws + (size_t)N_TOT * DIM;
    float* S2    = S1 + N_TOT;
    float* mu    = S2 + N_TOT;
    float* i2v   = mu + N_TOT;
    float* accum = i2v + N_TOT;                        // [0]=neg_sum, [1]=pos_sum

    k_normalize<<<N_TOT / 8, 256, 0, stream>>>(z, zn);
    k_init     <<<1, 32, 0, stream>>>(accum);
    k_stats    <<<N_TOT / ROWS, 256, 0, stream>>>(zn, S1, S2);
    k_stat_fin <<<N_TOT / 256, 256, 0, stream>>>(S1, S2, mu, i2v);
    k_neg      <<<N_TOT / ROWS, 256, 0, stream>>>(zn, labels, mu, i2v, accum);
    k_pos      <<<(N_TOT / 2) / 8, 256, 0, stream>>>(zn, labels, accum + 1);
    k_final    <<<1, 1, 0, stream>>>(accum, out);
}